// BertGTHead_37177236914708
// MI455X (gfx1250) — compile-verified
//
#include <hip/hip_runtime.h>
#include <math.h>

// Problem constants (from reference setup_inputs)
#define B_   16
#define S_   4096
#define H_   768
#define G_   16
#define WIN_ 15
#define WLEN (2*WIN_+1)        // 31
#define CH_  32                // S-chunks for streaming partials
#define SC_  (S_/CH_)          // 128 s-values per chunk
#define H4_  (H_/4)            // 192 float4 per row
#define K3H  (3*H_)            // 2304 feature length
#define NROW 272               // 16 cls rows + 256 gap rows

typedef float v2f __attribute__((ext_vector_type(2)));
typedef float v8f __attribute__((ext_vector_type(8)));

// ---------------------------------------------------------------------------
// Kernel 1: streaming partial reduction over sequence_output.
// grid = (CH_, B_), block = 192 threads, each thread owns 4 consecutive h.
// Writes per-(b,chunk) partial max / sum over masked s, plus mask count.
// ---------------------------------------------------------------------------
__global__ void k_partial(const float* __restrict__ seq,
                          const int* __restrict__ tt,
                          const int* __restrict__ wm,
                          float* __restrict__ pmax,
                          float* __restrict__ psum,
                          float* __restrict__ pcnt) {
    const int chunk = blockIdx.x;
    const int b     = blockIdx.y;
    const int tid   = threadIdx.x;            // 0..191
    __shared__ int smask[SC_];

    const int s0 = chunk * SC_;
    if (tid < SC_) {
        int idx = b * S_ + s0 + tid;
        smask[tid] = (tt[idx] == 0 && wm[idx] != 0) ? 1 : 0;
    }
    __syncthreads();

    const float4* seq4 = (const float4*)seq;
    const int base = (b * S_ + s0) * H4_ + tid;

    float mx = -INFINITY, my = -INFINITY, mz = -INFINITY, mw = -INFINITY;
    float sx = 0.f, sy = 0.f, sz = 0.f, sw = 0.f;

    for (int s = 0; s < SC_; ++s) {
        if (s + 8 < SC_)  // keep HBM pipe deep: gfx1250 global_prefetch_b8
            __builtin_prefetch((const void*)&seq4[base + (s + 8) * H4_], 0, 1);
        float4 v = seq4[base + s * H4_];
        if (smask[s]) {
            mx = fmaxf(mx, v.x); my = fmaxf(my, v.y);
            mz = fmaxf(mz, v.z); mw = fmaxf(mw, v.w);
            sx += v.x; sy += v.y; sz += v.z; sw += v.w;
        }
    }

    float4* pmax4 = (float4*)pmax;
    float4* psum4 = (float4*)psum;
    const int o = (b * CH_ + chunk) * H4_ + tid;
    pmax4[o] = make_float4(mx, my, mz, mw);
    psum4[o] = make_float4(sx, sy, sz, sw);

    if (tid == 0) {
        int c = 0;
        for (int s = 0; s < SC_; ++s) c += smask[s];
        pcnt[b * CH_ + chunk] = (float)c;
    }
}

// ---------------------------------------------------------------------------
// Kernel 2: combine chunk partials into text max/avg pools; write the 16 cls
// feature rows of F:  F[b] = [pooled | text_max_pool | text_avg_pool].
// grid = B_, block = 192.
// ---------------------------------------------------------------------------
__global__ void k_text(const float* __restrict__ pmax,
                       const float* __restrict__ psum,
                       const float* __restrict__ pcnt,
                       const float* __restrict__ pooled,
                       float* __restrict__ F) {
    const int b   = blockIdx.x;
    const int tid = threadIdx.x;              // 0..191

    const float4* pmax4 = (const float4*)pmax;
    const float4* psum4 = (const float4*)psum;

    float mx = -INFINITY, my = -INFINITY, mz = -INFINITY, mw = -INFINITY;
    float sx = 0.f, sy = 0.f, sz = 0.f, sw = 0.f;
    float cnt = 0.f;
    for (int ch = 0; ch < CH_; ++ch) {
        const int o = (b * CH_ + ch) * H4_ + tid;
        float4 m = pmax4[o];
        float4 s = psum4[o];
        mx = fmaxf(mx, m.x); my = fmaxf(my, m.y);
        mz = fmaxf(mz, m.z); mw = fmaxf(mw, m.w);
        sx += s.x; sy += s.y; sz += s.z; sw += s.w;
        cnt += pcnt[b * CH_ + ch];
    }
    const float inv = 1.0f / cnt;

    const float4* pooled4 = (const float4*)pooled;
    float4* F4 = (float4*)F;
    const int rb = b * (K3H / 4);             // row b, stride 576 float4
    F4[rb + tid]            = pooled4[b * H4_ + tid];
    F4[rb + H4_ + tid]      = make_float4(fmaxf(mx, 0.f), fmaxf(my, 0.f),
                                          fmaxf(mz, 0.f), fmaxf(mw, 0.f));
    F4[rb + 2 * H4_ + tid]  = make_float4(sx * inv, sy * inv, sz * inv, sw * inv);
}

// ---------------------------------------------------------------------------
// Kernel 3: per-(b,g) sliding-window pooling; writes the 256 gap feature rows:
// F[16 + b*16 + g] = [gap_token | window_max_pool | window_avg_pool].
// grid = B_*G_, block = 192.
// ---------------------------------------------------------------------------
__global__ void k_gap(const float* __restrict__ seq,
                      const int* __restrict__ tt,
                      const int* __restrict__ wm,
                      const int* __restrict__ gids,
                      float* __restrict__ F) {
    const int u   = blockIdx.x;
    const int b   = u >> 4;
    const int g   = u & 15;
    const int tid = threadIdx.x;              // 0..191
    __shared__ int wvalid[WLEN];

    const int gid = gids[b * G_ + g];
    if (tid < WLEN) {
        const int s = gid + tid - WIN_;
        int v = 0;
        if (s >= 0 && s < S_) {
            const int idx = b * S_ + s;
            v = (tt[idx] == 0 && wm[idx] != 0) ? 1 : 0;
        }
        wvalid[tid] = v;
    }
    __syncthreads();

    const float4* seq4 = (const float4*)seq;
    const float4 gv = seq4[(b * S_ + gid) * H4_ + tid];

    float mx = -INFINITY, my = -INFINITY, mz = -INFINITY, mw = -INFINITY;
    float sx = 0.f, sy = 0.f, sz = 0.f, sw = 0.f;
    int cnt = 0;
    for (int j = 0; j < WLEN; ++j) {
        if (wvalid[j]) {                      // uniform branch (LDS value)
            const int s = gid + j - WIN_;
            float4 v = seq4[(b * S_ + s) * H4_ + tid];
            mx = fmaxf(mx, v.x); my = fmaxf(my, v.y);
            mz = fmaxf(mz, v.z); mw = fmaxf(mw, v.w);
            sx += v.x; sy += v.y; sz += v.z; sw += v.w;
            ++cnt;
        }
    }
    const float inv = 1.0f / (float)cnt;

    float4* F4 = (float4*)F;
    const int row = 16 + b * 16 + g;
    const int rb  = row * (K3H / 4);
    F4[rb + tid]           = gv;
    F4[rb + H4_ + tid]     = make_float4(fmaxf(mx, 0.f), fmaxf(my, 0.f),
                                         fmaxf(mz, 0.f), fmaxf(mw, 0.f));
    F4[rb + 2 * H4_ + tid] = make_float4(sx * inv, sy * inv, sz * inv, sw * inv);
}

// ---------------------------------------------------------------------------
// Kernel 4: scores = F @ W + bias via V_WMMA_F32_16X16X4_F32 (exact fp32).
// 17 tiles of 16 rows: tile 0 = cls rows (cls_W/cls_b), tiles 1..16 = gap rows
// (gap_W/gap_b). grid = 17, block = 32 (one wave; EXEC all-ones for WMMA).
// A layout: lanes 0-15 hold K=k,k+1 for M=lane; lanes 16-31 hold K=k+2,k+3.
// B layout: weight value broadcast across all 16 N columns per K row.
// C layout: lane 0 -> rows 0-7 (N=0), lane 16 -> rows 8-15 (N=0).
// ---------------------------------------------------------------------------
__global__ void k_score(const float* __restrict__ F,
                        const float* __restrict__ gW,
                        const float* __restrict__ gb,
                        const float* __restrict__ cW,
                        const float* __restrict__ cb,
                        float* __restrict__ out) {
    const int tile = blockIdx.x;              // 0..16
    const int lane = threadIdx.x;             // 0..31
    const float* W   = (tile == 0) ? cW : gW;
    const float bias = (tile == 0) ? cb[0] : gb[0];
    const int rowBase = (tile == 0) ? 0 : 16 + (tile - 1) * 16;

    const int half = lane >> 4;               // 0: K=k,k+1   1: K=k+2,k+3
    const int m    = lane & 15;               // row within tile
    const float* Arow = F + (rowBase + m) * K3H;

    v8f c = {};
    for (int k = 0; k < K3H; k += 4) {
        const int ks = k + half * 2;
        v2f a;  a.x = Arow[ks]; a.y = Arow[ks + 1];
        v2f bb; bb.x = W[ks];   bb.y = W[ks + 1];
        c = __builtin_amdgcn_wmma_f32_16x16x4_f32(
                /*neg_a=*/false, a, /*neg_b=*/false, bb,
                /*c_mod=*/(short)0, c, /*reuse_a=*/false, /*reuse_b=*/false);
    }

    if (m == 0) {                             // lanes 0 and 16: N=0 column
        for (int i = 0; i < 8; ++i) {
            const int row = rowBase + half * 8 + i;
            const float val = c[i] + bias;
            int oi;
            if (row < 16) {
                oi = row * 17;                // cls score slot
            } else {
                const int r = row - 16;
                oi = (r >> 4) * 17 + 1 + (r & 15);
            }
            out[oi] = val;
        }
    }
}

// ---------------------------------------------------------------------------
extern "C" void kernel_launch(void* const* d_in, const int* in_sizes, int n_in,
                              void* d_out, int out_size, void* d_ws, size_t ws_size,
                              hipStream_t stream) {
    const float* seq    = (const float*)d_in[0];
    const float* pooled = (const float*)d_in[1];
    const int*   tt     = (const int*)d_in[2];
    const int*   wm     = (const int*)d_in[3];
    const int*   gids   = (const int*)d_in[4];
    const float* gW     = (const float*)d_in[5];
    const float* gb     = (const float*)d_in[6];
    const float* cW     = (const float*)d_in[7];
    const float* cb     = (const float*)d_in[8];
    float* out = (float*)d_out;

    // Workspace layout (floats): pmax | psum | pcnt | F
    float* pmax = (float*)d_ws;                          // B*CH*H = 393216
    float* psum = pmax + (size_t)B_ * CH_ * H_;          // 393216
    float* pcnt = psum + (size_t)B_ * CH_ * H_;          // 512
    float* F    = pcnt + (size_t)B_ * CH_;               // 272*2304 = 626688

    k_partial<<<dim3(CH_, B_), 192, 0, stream>>>(seq, tt, wm, pmax, psum, pcnt);
    k_text   <<<B_,            192, 0, stream>>>(pmax, psum, pcnt, pooled, F);
    k_gap    <<<B_ * G_,       192, 0, stream>>>(seq, tt, wm, gids, F);
    k_score  <<<17,            32,  0, stream>>>(F, gW, gb, cW, cb, out);
}